// SpatialAttention_22462678958146
// MI455X (gfx1250) — compile-verified
//
#include <hip/hip_runtime.h>
#include <math.h>

// Problem constants (from the reference).
#define B_    32
#define H_    112
#define W_    112
#define C_    192
#define PH    (H_ + 8)                // padded pooled height (kh reaches 7 for pad taps)
#define PW    (W_ + 8)                // padded pooled width
#define NPIX  (B_ * H_ * W_)          // 401,408 pixels
#define NSEG  (NPIX / 16)             // 25,088 16-pixel row segments (W=112 = 7*16)

typedef __attribute__((ext_vector_type(2))) float v2f;
typedef __attribute__((ext_vector_type(4))) float v4f;
typedef __attribute__((ext_vector_type(8))) float v8f;

// ---------------------------------------------------------------------------
// Pass 0: zero the padded pooled map so the conv halo reads zeros (SAME pad).
// 32*120*120*2 floats = 230,400 float4 -> 900 blocks exactly.
// ---------------------------------------------------------------------------
__global__ __launch_bounds__(256) void sa_zero_kernel(v4f* __restrict__ p) {
  const size_t i = (size_t)blockIdx.x * 256u + threadIdx.x;
  v4f z = {};
  p[i] = z;                       // regular store: pooled is re-read from L2 soon
}

// ---------------------------------------------------------------------------
// Pass 1: per-pixel channel mean + max into the padded pooled map.
// One wave32 per pixel; 192 channels = 6 fully coalesced 128B loads per wave.
// Non-temporal loads: x is a 308MB stream we do not want resident in L2.
// ---------------------------------------------------------------------------
__global__ __launch_bounds__(256) void sa_reduce_kernel(
    const float* __restrict__ x, v2f* __restrict__ pooled) {
  const int pix  = (int)((blockIdx.x * 256u + threadIdx.x) >> 5);  // wave id == pixel
  const int lane = (int)(threadIdx.x & 31u);
  const float* px = x + (size_t)pix * C_;
  float s = 0.0f, m = -INFINITY;
#pragma unroll
  for (int i = 0; i < C_ / 32; ++i) {
    float v = __builtin_nontemporal_load(px + lane + i * 32);
    s += v;
    m = fmaxf(m, v);
  }
#pragma unroll
  for (int off = 16; off > 0; off >>= 1) {
    s += __shfl_xor(s, off, 32);
    m = fmaxf(m, __shfl_xor(m, off, 32));
  }
  if (lane == 0) {
    const int b = pix / (H_ * W_);
    const int r = pix % (H_ * W_);
    const int h = r / W_;
    const int w = r % W_;
    v2f o; o.x = s * (1.0f / C_); o.y = m;
    pooled[((size_t)b * PH + (h + 3)) * PW + (w + 3)] = o;   // interior of halo
  }
}

// ---------------------------------------------------------------------------
// Pass 2: 7x7x2 conv + sigmoid via V_WMMA_F32_16X16X4_F32 (exact fp32 math).
// One wave per 16-pixel segment along W.  im2col tap t = kh*14 + kw*2 + c,
// 98 taps padded to 100 -> 25 chained K=4 WMMAs accumulating in C.
// A[m][k] = weight[t] broadcast over m  => D[m][n] = conv(out pixel n) for all m.
// Operand layout (32-bit, wave32): lanes 0-15 carry K=0,1 in VGPR0/1,
// lanes 16-31 carry K=2,3 — same half->K mapping used for A and B.
// Key trick: each lane's two taps per chunk are always the consecutive
// {avg,max} pair of one padded-pooled pixel -> a single aligned b64 gather,
// branch-free (halo supplies the zeros), so EXEC stays all-ones throughout.
// Taps 98/99 carry zero weights in A, so their B values contribute 0*finite=0.
// ---------------------------------------------------------------------------
__global__ __launch_bounds__(256) void sa_conv_kernel(
    const float* __restrict__ w98, const v2f* __restrict__ pooled,
    float* __restrict__ att) {
  __shared__ __align__(8) float wl[100];
  const int tid = (int)threadIdx.x;
  if (tid < 100) wl[tid] = (tid < 98) ? w98[tid] : 0.0f;
  __syncthreads();

  const int lane = tid & 31;
  const int gw   = (int)blockIdx.x * 8 + (tid >> 5);   // segment id, exact grid
  const int b    = gw / (H_ * 7);
  const int rem  = gw % (H_ * 7);
  const int h    = rem / 7;
  const int w0   = (rem % 7) * 16;
  const int n    = lane & 15;        // output pixel within segment (N dim)
  const int half = lane >> 4;        // 0: K=0,1   1: K=2,3

  // Padded base for conceptual (h-3, w0+n-3): padded index (h, w0+n).
  const v2f* base = pooled + ((size_t)b * PH + h) * PW + (w0 + n);

  v8f acc = {};
#pragma unroll
  for (int chunk = 0; chunk < 25; ++chunk) {
    const int t0  = chunk * 4;          // compile-time
    const int kh0 = t0 / 14;            // compile-time
    const int r0  = t0 % 14;            // compile-time, always even
    const int rp  = r0 + 2 * half;      // this lane's tap pair (even)
    const int cy  = (rp >= 14) ? 1 : 0; // carry only when r0==12 && half==1
    const int kh  = kh0 + cy;
    const int kw  = (rp - 14 * cy) >> 1;
    const v2f aw  = *(const v2f*)&wl[t0 + 2 * half];   // ds_load_b64
    const v2f bv  = base[kh * PW + kw];                // global_load_b64, no bounds check
    acc = __builtin_amdgcn_wmma_f32_16x16x4_f32(
        /*neg_a=*/false, aw, /*neg_b=*/false, bv,
        /*c_mod=*/(short)0, acc, /*reuse_a=*/false, /*reuse_b=*/false);
  }

  // D[0][n] lives in VGPR0 of lanes 0-15.
  if (half == 0) {
    att[((size_t)b * H_ + h) * W_ + w0 + n] = 1.0f / (1.0f + __expf(-acc[0]));
  }
}

// ---------------------------------------------------------------------------
// Pass 3: out = x * att (broadcast over 192 channels).  float4 streaming,
// non-temporal on the 308MB x/out streams; att (1.6MB) stays hot in L2.
// ---------------------------------------------------------------------------
__global__ __launch_bounds__(256) void sa_apply_kernel(
    const v4f* __restrict__ x4, const float* __restrict__ att,
    v4f* __restrict__ out4) {
  const size_t i = (size_t)blockIdx.x * 256u + threadIdx.x;   // exact grid
  v4f v = __builtin_nontemporal_load(&x4[i]);
  const float a = att[i / (C_ / 4)];                          // 48 float4 / pixel
  v *= a;
  __builtin_nontemporal_store(v, &out4[i]);
}

// ---------------------------------------------------------------------------
extern "C" void kernel_launch(void* const* d_in, const int* in_sizes, int n_in,
                              void* d_out, int out_size, void* d_ws, size_t ws_size,
                              hipStream_t stream) {
  const float* x      = (const float*)d_in[0];   // [B,H,W,C] fp32
  const float* conv_w = (const float*)d_in[1];   // [7,7,2,1] fp32 (98 floats)
  float*       out    = (float*)d_out;

  v2f*   pooled = (v2f*)d_ws;                          // [B][120][120] float2 = 3.7 MB
  float* att    = (float*)d_ws + (size_t)B_ * PH * PW * 2;  // [NPIX] = 1.6 MB

  // Pass 0: zero padded pooled map (230,400 float4 / 256 = 900 blocks, exact).
  sa_zero_kernel<<<(B_ * PH * PW * 2 / 4) / 256, 256, 0, stream>>>((v4f*)pooled);

  // Pass 1: 401,408 waves, 8 waves/block -> 50,176 blocks (exact).
  sa_reduce_kernel<<<NPIX / 8, 256, 0, stream>>>(x, pooled);

  // Pass 2: 25,088 segments, 8 waves/block -> 3,136 blocks (exact).
  sa_conv_kernel<<<NSEG / 8, 256, 0, stream>>>(conv_w, pooled, att);

  // Pass 3: 19,267,584 float4, 256 threads/block -> 75,264 blocks (exact).
  const int n4blocks = (NPIX * (C_ / 4)) / 256;
  sa_apply_kernel<<<n4blocks, 256, 0, stream>>>((const v4f*)x, att, (v4f*)out);
}